// HaarIDWT_7825430413852
// MI455X (gfx1250) — compile-verified
//
#include <hip/hip_runtime.h>

// ---------------------------------------------------------------------------
// Fused 3-level inverse Haar DWT, single pass over HBM.
//   out[8f+0..7] = (((lo[f] ± h2[f])*c ± h1[2f+..])*c ± h0[4f+..])*c
// Streaming via CDNA5 async global->LDS copies (ASYNCcnt), double-buffered.
// Non-temporal policy on both loads and stores: pure one-shot stream,
// 536 MB total traffic -> ~23 us at 23.3 TB/s; nothing is ever re-read.
// ---------------------------------------------------------------------------

typedef float v4f __attribute__((ext_vector_type(4)));

#define TILE_F4 256          // float4-groups of `lo` per tile (1 per thread)
#define TPB     4            // tiles per block

struct TileBuf {
    float4 lo[TILE_F4];        //  4 KB
    float4 h2[TILE_F4];        //  4 KB
    float4 h1[2 * TILE_F4];    //  8 KB
    float4 h0[4 * TILE_F4];    // 16 KB
};                             // 32 KB total; x2 buffers = 64 KB LDS

__device__ __forceinline__ void async_copy_b128(void* lds_dst, const float4* gsrc) {
    // LDS byte offset = low 32 bits of the flat shared-aperture address.
    unsigned lds = (unsigned)(unsigned long long)lds_dst;
    unsigned long long ga = (unsigned long long)gsrc;
    asm volatile("global_load_async_to_lds_b128 %0, %1, off th:TH_LOAD_NT"
                 :: "v"(lds), "v"(ga) : "memory");
}

__device__ __forceinline__ void issue_tile(const float4* __restrict__ lo,
                                           const float4* __restrict__ h2,
                                           const float4* __restrict__ h1,
                                           const float4* __restrict__ h0,
                                           TileBuf* b, long long G, int tid) {
    // 8 async B128 instructions per wave per tile (ASYNCcnt += 8).
    async_copy_b128(&b->lo[tid],                 lo + (G + tid));
    async_copy_b128(&b->h2[tid],                 h2 + (G + tid));
    async_copy_b128(&b->h1[tid],                 h1 + (2 * G + tid));
    async_copy_b128(&b->h1[tid + TILE_F4],       h1 + (2 * G + TILE_F4 + tid));
    async_copy_b128(&b->h0[tid],                 h0 + (4 * G + tid));
    async_copy_b128(&b->h0[tid + 1 * TILE_F4],   h0 + (4 * G + 1 * TILE_F4 + tid));
    async_copy_b128(&b->h0[tid + 2 * TILE_F4],   h0 + (4 * G + 2 * TILE_F4 + tid));
    async_copy_b128(&b->h0[tid + 3 * TILE_F4],   h0 + (4 * G + 3 * TILE_F4 + tid));
}

__device__ __forceinline__ void compute_tile(const TileBuf* b,
                                             v4f* __restrict__ out,
                                             long long G, int tid) {
    const float c = 0.70710678118654752440f;  // 1/sqrt(2)

    float4 lo = b->lo[tid];
    float4 h2 = b->h2[tid];
    float4 hA = b->h1[2 * tid];
    float4 hB = b->h1[2 * tid + 1];
    float lov[4] = {lo.x, lo.y, lo.z, lo.w};
    float h2v[4] = {h2.x, h2.y, h2.z, h2.w};
    float h1s[8] = {hA.x, hA.y, hA.z, hA.w, hB.x, hB.y, hB.z, hB.w};

    long long ob = 8 * (G + tid);
#pragma unroll
    for (int k = 0; k < 4; ++k) {
        // Level 3 (coarsest): x2[2t]=(lo+h2)*c, x2[2t+1]=(lo-h2)*c
        float x2e = (lov[k] + h2v[k]) * c;
        float x2o = (lov[k] - h2v[k]) * c;
        // Level 2
        float x10 = (x2e + h1s[2 * k]) * c;
        float x11 = (x2e - h1s[2 * k]) * c;
        float x12 = (x2o + h1s[2 * k + 1]) * c;
        float x13 = (x2o - h1s[2 * k + 1]) * c;
        // Level 1 (finest)
        float4 g = b->h0[4 * tid + k];
        v4f o0, o1;
        o0.x = (x10 + g.x) * c;  o0.y = (x10 - g.x) * c;
        o0.z = (x11 + g.y) * c;  o0.w = (x11 - g.y) * c;
        o1.x = (x12 + g.z) * c;  o1.y = (x12 - g.z) * c;
        o1.z = (x13 + g.w) * c;  o1.w = (x13 - g.w) * c;
        // Write-once stream: non-temporal 128-bit stores.
        __builtin_nontemporal_store(o0, out + (ob + 2 * k));
        __builtin_nontemporal_store(o1, out + (ob + 2 * k + 1));
    }
}

__global__ void __launch_bounds__(256)
haar_idwt3_kernel(const float4* __restrict__ lo, const float4* __restrict__ h2,
                  const float4* __restrict__ h1, const float4* __restrict__ h0,
                  v4f* __restrict__ out, int numTiles) {
    __shared__ TileBuf sbuf[2];
    const int tid = threadIdx.x;

    long long baseTile = (long long)blockIdx.x * TPB;
    if (baseTile >= numTiles) return;
    int nt = (numTiles - baseTile < TPB) ? (int)(numTiles - baseTile) : TPB;

    issue_tile(lo, h2, h1, h0, &sbuf[0], baseTile * TILE_F4, tid);

    for (int i = 0; i < nt; ++i) {
        if (i + 1 < nt) {
            // Prefetch next tile into the other buffer (safe: barrier at end
            // of previous iteration guarantees nobody still reads it).
            issue_tile(lo, h2, h1, h0, &sbuf[(i + 1) & 1],
                       (baseTile + i + 1) * TILE_F4, tid);
            // 8 loads of tile i+1 in flight; <=8 means tile i fully landed
            // (async loads complete in order within a wave).
            asm volatile("s_wait_asynccnt 0x8" ::: "memory");
        } else {
            asm volatile("s_wait_asynccnt 0x0" ::: "memory");
        }
        __syncthreads();                       // tile i visible block-wide
        compute_tile(&sbuf[i & 1], out, (baseTile + i) * TILE_F4, tid);
        __syncthreads();                       // done reading before refill
    }
}

extern "C" void kernel_launch(void* const* d_in, const int* in_sizes, int n_in,
                              void* d_out, int out_size, void* d_ws, size_t ws_size,
                              hipStream_t stream) {
    // setup_inputs order: low_last, high0 (finest), high1, high2 (coarsest)
    const float4* lo = (const float4*)d_in[0];
    const float4* h0 = (const float4*)d_in[1];
    const float4* h1 = (const float4*)d_in[2];
    const float4* h2 = (const float4*)d_in[3];
    v4f* out = (v4f*)d_out;

    long long F4 = (long long)in_sizes[0] / 4;        // float4 groups of lo
    int numTiles = (int)(F4 / TILE_F4);               // 8192 for B=16,C=128,L=4096
    int blocks = (numTiles + TPB - 1) / TPB;          // 2048

    haar_idwt3_kernel<<<blocks, 256, 0, stream>>>(lo, h2, h1, h0, out, numTiles);
}